// Entangle_layer_41369124995658
// MI455X (gfx1250) — compile-verified
//
#include <hip/hip_runtime.h>

// ---------------------------------------------------------------------------
// Entangle layer: 128 batches x 8 blocks x 2^14 complex state.
// Memory bound: 128 MB read + 256 MB written  ->  ~16.5 us @ 23.3 TB/s.
// Data path: TDM (tensor_load_to_lds) stages each half-vector into LDS,
// waves apply the 2x2 complex gates out of LDS, results streamed with
// non-temporal stores (output is write-once, keep it out of L2).
// ---------------------------------------------------------------------------

typedef unsigned int       u32;
typedef unsigned long long u64;
typedef __attribute__((ext_vector_type(2))) float        f2;
typedef __attribute__((ext_vector_type(4))) unsigned int v4u;
typedef __attribute__((ext_vector_type(4))) int          v4i;
typedef __attribute__((ext_vector_type(8))) int          v8i;

#define N_STATE 16384
#define HALF    8192

__device__ __forceinline__ f2 mk(float x, float y) { f2 r; r.x = x; r.y = y; return r; }

// Apply one gate over a (bit-clear, bit-set) pair, for both output copies.
// type: 0 = none, 1 = control (diag(1,-i) lo / diag(1,+i) hi),
//       2 = target (HSH lo / HSZH hi).
__device__ __forceinline__ void apply_gate(int type, f2& lu, f2& lw, f2& gu, f2& gw) {
    if (type == 1) {
        lw = mk(lw.y, -lw.x);          // * -i
        gw = mk(-gw.y, gw.x);          // * +i
    } else if (type == 2) {
        f2 u = lu, w = lw;             // lo: a=(1-i)/2, b=(1+i)/2
        lu = mk(0.5f * (u.x + u.y + w.x - w.y), 0.5f * (u.y - u.x + w.x + w.y));
        lw = mk(0.5f * (u.x - u.y + w.x + w.y), 0.5f * (u.x + u.y + w.y - w.x));
        u = gu; w = gw;                // hi: a<->b swapped
        gu = mk(0.5f * (u.x - u.y + w.x + w.y), 0.5f * (u.x + u.y + w.y - w.x));
        gw = mk(0.5f * (u.x + u.y + w.x - w.y), 0.5f * (u.y - u.x + w.x + w.y));
    }
}

// Issue one TDM 2-D copy (rows of fp32) into LDS.  D# bitfields per
// CDNA5 ISA (group0 128b, group1 256b).
__device__ __forceinline__ void tdm_load_2d(u32 lds_addr, const float* gptr,
                                            u32 tile0, u32 tile1,
                                            u32 td0, u32 td1, u64 stride0) {
    u64 ga = (u64)(size_t)gptr;
    v4u g0;
    g0[0] = 1u;                                            // count=1, user D#
    g0[1] = lds_addr;                                      // lds_addr[31:0]
    g0[2] = (u32)ga;                                       // global_addr[31:0]
    g0[3] = (u32)((ga >> 32) & 0x01FFFFFFull) | (2u << 30); // addr[56:32] | type=2
    v8i g1;
    g1[0] = (int)(2u << 16);                               // data_size=2 (4B)
    g1[1] = (int)((td0 & 0xFFFFu) << 16);                  // tensor_dim0[15:0]
    g1[2] = (int)(((td0 >> 16) & 0xFFFFu) | ((td1 & 0xFFFFu) << 16));
    g1[3] = (int)(((td1 >> 16) & 0xFFFFu) | ((tile0 & 0xFFFFu) << 16));
    g1[4] = (int)(tile1 & 0xFFFFu);                        // tile_dim1, tile_dim2=0
    g1[5] = (int)(u32)(stride0 & 0xFFFFFFFFull);           // tensor_dim0_stride lo
    g1[6] = (int)(u32)((stride0 >> 32) & 0xFFFFull);       // stride hi | dim1_stride lo
    g1[7] = 0;
    v4i z4 = {0, 0, 0, 0};
#if __clang_major__ >= 23
    v8i z8 = {0, 0, 0, 0, 0, 0, 0, 0};
    __builtin_amdgcn_tensor_load_to_lds(g0, g1, z4, z4, z8, 0);
#else
    __builtin_amdgcn_tensor_load_to_lds(g0, g1, z4, z4, 0);
#endif
}

__global__ __launch_bounds__(256) void entangle_kernel(const float* __restrict__ re,
                                                       const float* __restrict__ im,
                                                       f2* __restrict__ out) {
    __shared__ float lds_re[HALF];
    __shared__ float lds_im[HALF];

    const u32 wg  = blockIdx.x;        // b*16 + blk*2 + half
    const u32 b   = wg >> 4;
    const u32 blk = (wg >> 1) & 7u;
    const u32 hh  = wg & 1u;

    // Per-block gate metadata in LDS-index space (blk7's bit13 -> bit12; blk0's
    // bit13 control is uniform over the half -> handled as a global phase).
    const u32 mLowArr[8]  = {1u << 0, 1u << 8, 1u << 7,  1u << 0, 1u << 6,  1u << 5,  1u << 1, 1u << 2};
    const int tLowArr[8]  = {1,       2,       2,        2,       1,        1,        2,       2};
    const u32 mHighArr[8] = {1u << 1, 1u << 9, 1u << 10, 1u << 4, 1u << 12, 1u << 11, 1u << 3, 1u << 12};
    const int tHighArr[8] = {0,       1,       1,        2,       1,        2,        1,       2};

    const u32 mLow  = mLowArr[blk];
    const u32 mHigh = mHighArr[blk];
    const int tLow  = tLowArr[blk];
    const int tHigh = tHighArr[blk];
    const bool uniPhase = (blk == 0) && (hh == 1u);   // blk0 control at bit13
    const bool isB7     = (blk == 7);                 // splits on bit12

    const float* src_re = re + (size_t)(b * 8u + blk) * N_STATE;
    const float* src_im = im + (size_t)(b * 8u + blk) * N_STATE;

    if (threadIdx.x < 32) {            // one wave drives the TDM
        u32 a_re = (u32)(size_t)(&lds_re[0]);
        u32 a_im = (u32)(size_t)(&lds_im[0]);
        u32 goff   = isB7 ? (hh * 4096u) : (hh * 8192u);
        u64 stride = isB7 ? 8192u : 4096u;   // blk7: rows {h*4096, 8192+h*4096}
        tdm_load_2d(a_re, src_re + goff, 4096u, 2u, 4096u, 2u, stride);
        tdm_load_2d(a_im, src_im + goff, 4096u, 2u, 4096u, 2u, stride);
        __builtin_amdgcn_s_wait_tensorcnt(0);
    }
    __syncthreads();

    const size_t obase = (size_t)(b * 8u + blk) * 2u * N_STATE;
    f2* oLo = out + obase;
    f2* oHi = out + obase + N_STATE;

    for (u32 it = 0; it < 8; ++it) {
        u32 s = threadIdx.x + (it << 8);                       // 0..2047 subcubes
        u32 e    = ((s & ~(mLow  - 1u)) << 1) | (s & (mLow  - 1u));
        u32 base = ((e & ~(mHigh - 1u)) << 1) | (e & (mHigh - 1u));
        u32 i00 = base, i01 = base | mLow, i10 = base | mHigh, i11 = base | mLow | mHigh;

        f2 l00 = mk(lds_re[i00], lds_im[i00]);
        f2 l01 = mk(lds_re[i01], lds_im[i01]);
        f2 l10 = mk(lds_re[i10], lds_im[i10]);
        f2 l11 = mk(lds_re[i11], lds_im[i11]);
        f2 g00 = l00, g01 = l01, g10 = l10, g11 = l11;

        apply_gate(tLow,  l00, l01, g00, g01);
        apply_gate(tLow,  l10, l11, g10, g11);
        apply_gate(tHigh, l00, l10, g00, g10);
        apply_gate(tHigh, l01, l11, g01, g11);

        if (uniPhase) {   // blk0, half with bit13 set: lo *= -i, hi *= +i
            l00 = mk(l00.y, -l00.x); l01 = mk(l01.y, -l01.x);
            l10 = mk(l10.y, -l10.x); l11 = mk(l11.y, -l11.x);
            g00 = mk(-g00.y, g00.x); g01 = mk(-g01.y, g01.x);
            g10 = mk(-g10.y, g10.x); g11 = mk(-g11.y, g11.x);
        }

        u32 n00, n01, n10, n11;
        if (isB7) {
            n00 = hh * 4096u + (i00 & 4095u) + ((i00 >> 12) << 13);
            n01 = hh * 4096u + (i01 & 4095u) + ((i01 >> 12) << 13);
            n10 = hh * 4096u + (i10 & 4095u) + ((i10 >> 12) << 13);
            n11 = hh * 4096u + (i11 & 4095u) + ((i11 >> 12) << 13);
        } else {
            u32 off = hh * 8192u;
            n00 = off + i00; n01 = off + i01; n10 = off + i10; n11 = off + i11;
        }

        __builtin_nontemporal_store(l00, oLo + n00);
        __builtin_nontemporal_store(l01, oLo + n01);
        __builtin_nontemporal_store(l10, oLo + n10);
        __builtin_nontemporal_store(l11, oLo + n11);
        __builtin_nontemporal_store(g00, oHi + n00);
        __builtin_nontemporal_store(g01, oHi + n01);
        __builtin_nontemporal_store(g10, oHi + n10);
        __builtin_nontemporal_store(g11, oHi + n11);
    }
}

extern "C" void kernel_launch(void* const* d_in, const int* in_sizes, int n_in,
                              void* d_out, int out_size, void* d_ws, size_t ws_size,
                              hipStream_t stream) {
    const float* re = (const float*)d_in[0];
    const float* im = (const float*)d_in[1];
    // 128 batches * 8 blocks * 2 halves = 2048 workgroups, 256 threads each.
    entangle_kernel<<<dim3(2048), dim3(256), 0, stream>>>(re, im, (f2*)d_out);
}